// ImportanceRaySampler_3289944949523
// MI455X (gfx1250) — compile-verified
//
#include <hip/hip_runtime.h>
#include <hip/hip_bf16.h>
#include <stdint.h>

// ImportanceRaySampler fused one-pass kernel for gfx1250 (MI455X).
// B=4, R=65536, C=64 coarse bins, NS=128 fine samples.
// One wave32 per ray, 4 rays per 128-thread block (per-wave LDS regions,
// no inter-wave communication -> no s_barrier; wave-synchronous ordering).
// Memory floor: ~536 MB traffic -> ~23 us @ 23.3 TB/s. Everything fused.

#define C_BINS 64
#define NSAMP  128
#define ALPHA_F 1e-5f
#define WPB 4   // waves (rays) per block

// DPP8 selector for intra-8-lane XOR permute: sel[i] = i ^ j
#define DPP8_XOR(j) ( ((0^(j))    ) | ((1^(j))<< 3) | ((2^(j))<< 6) | ((3^(j))<< 9) \
                    | ((4^(j))<<12) | ((5^(j))<<15) | ((6^(j))<<18) | ((7^(j))<<21) )

// Cross-lane XOR shuffle. jl in {1,2,4}: VALU v_mov_b32_dpp8 (no DS unit,
// no dscnt wait). jl in {8,16}: ds_swizzle group-of-32 XOR.
__device__ __forceinline__ float lane_xor(float x, int jl) {
    int xi = __float_as_int(x);
    int r;
    switch (jl) {
        case 1:  r = __builtin_amdgcn_mov_dpp8(xi, DPP8_XOR(1)); break;
        case 2:  r = __builtin_amdgcn_mov_dpp8(xi, DPP8_XOR(2)); break;
        case 4:  r = __builtin_amdgcn_mov_dpp8(xi, DPP8_XOR(4)); break;
        case 8:  r = __builtin_amdgcn_ds_swizzle(xi, (8  << 10) | 0x1f); break;
        default: r = __builtin_amdgcn_ds_swizzle(xi, (16 << 10) | 0x1f); break;
    }
    return __int_as_float(r);
}

// x + shfl_up(x, d) within rows of 16 via DPP row_shr:d; bound lanes add 0.
// CTRL must be a template parameter so the builtin sees a true ICE.
// (Compiler folds this into a single v_add_f32_dpp.)
template <int CTRL>
__device__ __forceinline__ float dpp_row_shr_add(float x) {
    int y = __builtin_amdgcn_update_dpp(0, __float_as_int(x), CTRL,
                                        0xf, 0xf, true);
    return x + __int_as_float(y);
}

__global__ __launch_bounds__(32 * WPB) void importance_ray_sampler_kernel(
    const float* __restrict__ ts,      // [B*R, 64]
    const float* __restrict__ weights, // [B*R, 64]
    const float* __restrict__ t0,      // [B*R]
    const float* __restrict__ t1,      // [B*R]
    const float* __restrict__ u,       // [B*R, 128]
    const float* __restrict__ t_rand,  // [B*R, 128]
    float* __restrict__ out,           // [B*R, 128]
    int n_rays)
{
    // Per-wave LDS regions (no cross-wave sharing -> no block barriers).
    // s_tsp[w][2..65] = ts[0..63]; s_tsp[w][1], s_tsp[w][66] are mirror pads:
    // lower = .5*(tsp[i+1]+tsp[i+2]), upper = .5*(tsp[i+2]+tsp[i+3]) reproduce
    // the t0/t1 edge cases branch-free. Async b64 dest starts at s_tsp[w][2]
    // -> 8B-aligned per lane (row stride 72 floats keeps 8B alignment).
    __shared__ float s_tsp[WPB][72];
    __shared__ float s_cdf[WPB][C_BINS];

    const int lane = threadIdx.x & 31;
    // Force wave id into an SGPR so ray / base pointers / guard are scalar:
    // uniform s_cmp guard, SALU 64-bit address math, s_load for t0/t1, and
    // a direct SGPR-pair feed for the async-DMA "s" constraint.
    const int wave = __builtin_amdgcn_readfirstlane(threadIdx.x >> 5);
    const int ray  = (int)blockIdx.x * WPB + wave;   // SGPR-uniform
    if (ray >= n_rays) return;                       // uniform branch

    const size_t rowC = (size_t)ray << 6;  // *64
    const size_t rowS = (size_t)ray << 7;  // *128
    const float* ts_ptr = ts      + rowC;
    const float* w_ptr  = weights + rowC;
    const float* u_ptr  = u       + rowS;
    const float* tr_ptr = t_rand  + rowS;
    float*       o_ptr  = out     + rowS;

    // ---- Stage 1: async DMA ts row (256B) into LDS, 8B per lane ----------
    {
        unsigned lds_addr = (unsigned)(size_t)(&s_tsp[wave][2]) + 8u * (unsigned)lane;
        unsigned voff     = 8u * (unsigned)lane;
        unsigned long long base = (unsigned long long)(size_t)ts_ptr;
        asm volatile("global_load_async_to_lds_b64 %0, %1, %2"
                     :: "v"(lds_addr), "v"(voff), "s"(base)
                     : "memory");
    }

    // ---- Stage 2: weights -> unnormalized cdf in LDS (overlaps the DMA) ---
    float2 wv = reinterpret_cast<const float2*>(w_ptr)[lane]; // elems 2l,2l+1
    float w0 = wv.x + ALPHA_F;
    float w1 = wv.y + ALPHA_F;
    float pairsum = w0 + w1;

    // Wave-inclusive scan of per-lane pair sums, VALU-only:
    // v_add_f32_dpp row_shr chain within rows of 16, then one readlane(15)
    // broadcast added to the upper half. No DS, no dscnt waits.
    float scan = pairsum;
    scan = dpp_row_shr_add<0x111>(scan); // row_shr:1
    scan = dpp_row_shr_add<0x112>(scan); // row_shr:2
    scan = dpp_row_shr_add<0x114>(scan); // row_shr:4
    scan = dpp_row_shr_add<0x118>(scan); // row_shr:8
    float s15 = __int_as_float(
        __builtin_amdgcn_readlane(__float_as_int(scan), 15));
    scan += (lane >= 16) ? s15 : 0.0f;
    float total = __int_as_float(
        __builtin_amdgcn_readlane(__float_as_int(scan), 31)); // SGPR-uniform

    *reinterpret_cast<float2*>(&s_cdf[wave][2 * lane]) =
        make_float2(scan - w1, scan); // single ds_store_b64

    float t0v = t0[ray];   // uniform address -> s_load
    float t1v = t1[ray];

    // ---- Stage 3: wait for DMA, patch mirror pads -------------------------
    asm volatile("s_wait_asynccnt 0" ::: "memory");
    __builtin_amdgcn_wave_barrier();   // codegen ordering; wave is lockstep
    if (lane == 0) {
        s_tsp[wave][1]  = 2.0f * t0v - s_tsp[wave][2];  // lower[0]  == t0
        s_tsp[wave][66] = 2.0f * t1v - s_tsp[wave][65]; // upper[63] == t1
    }
    __builtin_amdgcn_wave_barrier();

    // ---- Stage 4: 4 interleaved binary searches (searchsorted 'left') -----
    // All 4 loads of a level are independent -> 4 ds_load_b32 per s_wait.
    float4 uv = reinterpret_cast<const float4*>(u_ptr)[lane];
    float4 tr = reinterpret_cast<const float4*>(tr_ptr)[lane];
    float key[4] = { uv.x * total, uv.y * total, uv.z * total, uv.w * total };
    float trf[4] = { tr.x, tr.y, tr.z, tr.w };

    int pos[4] = { 0, 0, 0, 0 };
#pragma unroll
    for (int off = 32; off >= 1; off >>= 1) {
        float c[4];
#pragma unroll
        for (int q = 0; q < 4; ++q)
            c[q] = s_cdf[wave][pos[q] + off - 1];   // 4 independent LDS loads
#pragma unroll
        for (int q = 0; q < 4; ++q)
            pos[q] += (c[q] < key[q]) ? off : 0;
    }

    // ---- Stage 5: batched gather (12 independent LDS loads) + lerp --------
    float ga[4], gb[4], gc[4];
#pragma unroll
    for (int q = 0; q < 4; ++q) {
        int ind = pos[q] > (C_BINS - 1) ? (C_BINS - 1) : pos[q];
        ga[q] = s_tsp[wave][ind + 1];
        gb[q] = s_tsp[wave][ind + 2];
        gc[q] = s_tsp[wave][ind + 3];
    }
    float v[4];
#pragma unroll
    for (int q = 0; q < 4; ++q) {
        float lower = 0.5f * (ga[q] + gb[q]);
        float upper = 0.5f * (gb[q] + gc[q]);
        v[q] = lower + (upper - lower) * trf[q];
    }

    // ---- Stage 6: in-register bitonic sort of 128 elems, e = 4*lane + q ---
#pragma unroll
    for (int k = 2; k <= NSAMP; k <<= 1) {
#pragma unroll
        for (int j = k >> 1; j >= 1; j >>= 1) {
            if (j <= 2) {
                // in-register pairs (q, q^j)
#pragma unroll
                for (int q = 0; q < 4; ++q) {
                    int p = q ^ j;
                    if (p > q) {
                        bool asc = (((4 * lane + q) & k) == 0);
                        float lo = fminf(v[q], v[p]);
                        float hi = fmaxf(v[q], v[p]);
                        v[q] = asc ? lo : hi;
                        v[p] = asc ? hi : lo;
                    }
                }
            } else {
                // cross-lane: DPP8 for xor{1,2,4}, ds_swizzle for xor{8,16}
                int jl = j >> 2;
#pragma unroll
                for (int q = 0; q < 4; ++q) {
                    float partner = lane_xor(v[q], jl);
                    bool asc     = (((4 * lane + q) & k) == 0);
                    bool lowhalf = ((lane & jl) == 0);
                    float lo = fminf(v[q], partner);
                    float hi = fmaxf(v[q], partner);
                    v[q] = (lowhalf == asc) ? lo : hi;
                }
            }
        }
    }

    // ---- Stage 7: coalesced float4 store ----------------------------------
    reinterpret_cast<float4*>(o_ptr)[lane] = make_float4(v[0], v[1], v[2], v[3]);
}

extern "C" void kernel_launch(void* const* d_in, const int* in_sizes, int n_in,
                              void* d_out, int out_size, void* d_ws, size_t ws_size,
                              hipStream_t stream) {
    const float* ts      = (const float*)d_in[0];
    const float* weights = (const float*)d_in[1];
    const float* t0      = (const float*)d_in[2];
    const float* t1      = (const float*)d_in[3];
    const float* u       = (const float*)d_in[4];
    const float* t_rand  = (const float*)d_in[5];
    float* out = (float*)d_out;

    const int n_rays = in_sizes[2]; // t0 is [B, R, 1] -> B*R rays
    const int blocks = (n_rays + WPB - 1) / WPB;

    importance_ray_sampler_kernel<<<blocks, 32 * WPB, 0, stream>>>(
        ts, weights, t0, t1, u, t_rand, out, n_rays);
}